// SEAL_GNN_12120397709999
// MI455X (gfx1250) — compile-verified
//
#include <hip/hip_runtime.h>
#include <hip/hip_bf16.h>

typedef __attribute__((ext_vector_type(2))) float v2f;
typedef __attribute__((ext_vector_type(8))) float v8f;

#define CH 64  // hidden channels (H)

// ---------------------------------------------------------------- utilities
__global__ void k_fill(float* __restrict__ p, float v, int n) {
    int i = blockIdx.x * blockDim.x + threadIdx.x;
    if (i < n) p[i] = v;
}

__global__ void k_deg_accum(const int* __restrict__ dst, float* __restrict__ deg, int E) {
    int e = blockIdx.x * blockDim.x + threadIdx.x;
    if (e < E) atomicAdd(&deg[dst[e]], 1.0f);
}

__global__ void k_rsqrt_inplace(float* __restrict__ p, int n) {
    int i = blockIdx.x * blockDim.x + threadIdx.x;
    if (i < n) p[i] = rsqrtf(p[i]);   // deg >= 1 (self-loops), no clamp needed
}

// ----------------------------------------------------- WMMA f32 GEMM (N x K) @ (K x 64)
// One wave32 per 16x16 output tile. V_WMMA_F32_16X16X4_F32, K stepped by 4.
// A-fragment (16x4 f32): lane L -> row = L&15, holds K = 2*(L>>4) + {0,1}
// B-fragment (4x16 f32): lane L -> col = L&15, holds K = 2*(L>>4) + {0,1}
// C/D (16x16 f32): VGPR i -> row = i + 8*(L>>4), col = L&15
template <bool RELU>
__global__ __launch_bounds__(32)
void k_gemm_wmma(const float* __restrict__ A, const float* __restrict__ W,
                 float* __restrict__ out, int K) {
    const int mtile = blockIdx.x;
    const int ntile = blockIdx.y;
    const int lane  = threadIdx.x;
    const int r     = lane & 15;
    const int khalf = lane >> 4;        // 0 or 1
    const int row   = mtile * 16 + r;   // N % 16 == 0: no guard needed
    const int col   = ntile * 16 + r;

    const float* __restrict__ arow = A + (size_t)row * K;
    v8f acc = {};
    for (int k0 = 0; k0 < K; k0 += 4) {
        const int ka = k0 + khalf * 2;
        v2f a, b;
        a.x = arow[ka];
        a.y = arow[ka + 1];
        if (RELU) { a.x = fmaxf(a.x, 0.0f); a.y = fmaxf(a.y, 0.0f); }
        b.x = W[(size_t)ka * CH + col];
        b.y = W[(size_t)(ka + 1) * CH + col];
        acc = __builtin_amdgcn_wmma_f32_16x16x4_f32(
            /*neg_a=*/false, a, /*neg_b=*/false, b,
            /*c_mod=*/(short)0, acc, /*reuse_a=*/false, /*reuse_b=*/false);
    }
#pragma unroll
    for (int i = 0; i < 8; ++i) {
        const int orow = mtile * 16 + i + 8 * khalf;
        out[(size_t)orow * CH + col] = acc[i];
    }
}

// ------------------------------------------- self-loop + bias init of aggregation
__global__ void k_agg_init(const float* __restrict__ h, const float* __restrict__ dinv,
                           const float* __restrict__ bias, float* __restrict__ out, int n) {
    int t = blockIdx.x * blockDim.x + threadIdx.x;
    if (t < n * CH) {
        int i = t >> 6;          // node
        int c = t & (CH - 1);    // channel
        float d = dinv[i];
        out[t] = d * d * h[t] + bias[c];
    }
}

// ------------------------------------------- edge scatter: out[dst] += w * h[src]
__global__ void k_scatter(const int* __restrict__ src, const int* __restrict__ dst,
                          const float* __restrict__ dinv, const float* __restrict__ h,
                          float* __restrict__ out, int E) {
    int t = blockIdx.x * blockDim.x + threadIdx.x;
    if (t < E * CH) {
        int e = t >> 6;
        int c = t & (CH - 1);
        int s = src[e], d = dst[e];
        float w = dinv[s] * dinv[d];
        atomicAdd(&out[(size_t)d * CH + c], w * h[(size_t)s * CH + c]);
    }
}

// ------------------------------------------- mean-pool scatter + head
__global__ void k_pool_scatter(const float* __restrict__ h, const int* __restrict__ batch,
                               float* __restrict__ pool, float* __restrict__ cnt, int n) {
    int t = blockIdx.x * blockDim.x + threadIdx.x;
    if (t < n * CH) {
        int i = t >> 6;
        int c = t & (CH - 1);
        int g = batch[i];
        atomicAdd(&pool[(size_t)g * CH + c], h[t]);
        if (c == 0) atomicAdd(&cnt[g], 1.0f);
    }
}

__global__ __launch_bounds__(32)
void k_head(const float* __restrict__ pool, const float* __restrict__ cnt,
            const float* __restrict__ lw, const float* __restrict__ lb,
            float* __restrict__ out) {
    int g = blockIdx.x;
    int lane = threadIdx.x;                 // wave32
    const float* p = pool + (size_t)g * CH;
    float v = p[lane] * lw[lane] + p[lane + 32] * lw[lane + 32];
    for (int off = 16; off > 0; off >>= 1) v += __shfl_down(v, off, 32);
    if (lane == 0) out[g] = v / fmaxf(cnt[g], 1.0f) + lb[0];
}

// ---------------------------------------------------------------- launcher
extern "C" void kernel_launch(void* const* d_in, const int* in_sizes, int n_in,
                              void* d_out, int out_size, void* d_ws, size_t ws_size,
                              hipStream_t stream) {
    const float* x   = (const float*)d_in[0];
    const int*   ei  = (const int*)d_in[1];
    const int*   bat = (const int*)d_in[2];
    const float* W1  = (const float*)d_in[3];
    const float* b1  = (const float*)d_in[4];
    const float* W2  = (const float*)d_in[5];
    const float* b2  = (const float*)d_in[6];
    const float* W3  = (const float*)d_in[7];
    const float* b3  = (const float*)d_in[8];
    const float* lw  = (const float*)d_in[9];
    const float* lb  = (const float*)d_in[10];

    const int N  = in_sizes[2];         // 100000
    const int E  = in_sizes[1] / 2;     // 1600000
    const int IN = in_sizes[0] / N;     // 128
    const int G  = out_size;            // 512
    const int* srcv = ei;               // edge_index[0]
    const int* dstv = ei + E;           // edge_index[1]

    float* ws   = (float*)d_ws;
    float* dinv = ws;                               // N
    float* bufA = dinv + N;                         // N*CH
    float* bufB = bufA + (size_t)N * CH;            // N*CH
    float* pool = bufB + (size_t)N * CH;            // G*CH
    float* cnt  = pool + (size_t)G * CH;            // G

    const int T = 256;
    dim3 gemm_grid((N + 15) / 16, CH / 16);

    // degree -> dinv
    k_fill<<<(N + T - 1) / T, T, 0, stream>>>(dinv, 1.0f, N);          // self-loop
    k_deg_accum<<<(E + T - 1) / T, T, 0, stream>>>(dstv, dinv, E);
    k_rsqrt_inplace<<<(N + T - 1) / T, T, 0, stream>>>(dinv, N);

    const int nch = N * CH, ech = E * CH;

    // layer 1: h = x @ W1 ; o = D^-1/2 (A+I) D^-1/2 h + b1
    k_gemm_wmma<false><<<gemm_grid, 32, 0, stream>>>(x, W1, bufA, IN);
    k_agg_init<<<(nch + T - 1) / T, T, 0, stream>>>(bufA, dinv, b1, bufB, N);
    k_scatter<<<(ech + T - 1) / T, T, 0, stream>>>(srcv, dstv, dinv, bufA, bufB, E);

    // layer 2 (ReLU of layer-1 output fused into GEMM A-load)
    k_gemm_wmma<true><<<gemm_grid, 32, 0, stream>>>(bufB, W2, bufA, CH);
    k_agg_init<<<(nch + T - 1) / T, T, 0, stream>>>(bufA, dinv, b2, bufB, N);
    k_scatter<<<(ech + T - 1) / T, T, 0, stream>>>(srcv, dstv, dinv, bufA, bufB, E);

    // layer 3 (no ReLU after)
    k_gemm_wmma<true><<<gemm_grid, 32, 0, stream>>>(bufB, W3, bufA, CH);
    k_agg_init<<<(nch + T - 1) / T, T, 0, stream>>>(bufA, dinv, b3, bufB, N);
    k_scatter<<<(ech + T - 1) / T, T, 0, stream>>>(srcv, dstv, dinv, bufA, bufB, E);

    // global mean pool + linear head
    k_fill<<<(G * CH + T - 1) / T, T, 0, stream>>>(pool, 0.0f, G * CH);
    k_fill<<<(G + T - 1) / T, T, 0, stream>>>(cnt, 0.0f, G);
    k_pool_scatter<<<(nch + T - 1) / T, T, 0, stream>>>(bufB, bat, pool, cnt, N);
    k_head<<<G, 32, 0, stream>>>(pool, cnt, lw, lb, (float*)d_out);
}